// AGCN_max_med_fusion_61237643706585
// MI455X (gfx1250) — compile-verified
//
#include <hip/hip_runtime.h>

#define NSAMP 128
#define CCH   1024
#define HWSZ  196
#define HDIM  14
#define ODIM  256
#define KSEL  16
#define BN_EPS 1e-5f

typedef __attribute__((ext_vector_type(2))) float v2f;
typedef __attribute__((ext_vector_type(8))) float v8f;

// ---------------------------------------------------------------------------
// Kernel 1: fre = sum_c fpam[n,c,:], stable rank-sort of 196 values,
// emit scene indices (workspace) and rows/cols (output tail, as float).
// One block per sample, 256 threads (8 waves).
// ---------------------------------------------------------------------------
__global__ __launch_bounds__(256) void k_scene(const float* __restrict__ fpam,
                                               int* __restrict__ scene_ws,
                                               float* __restrict__ out)
{
    __shared__ float fre[HWSZ];
    __shared__ int   sorted[HWSZ];

    const int n = blockIdx.x;
    const int t = threadIdx.x;

    if (t < HWSZ) {
        const float* p = fpam + (size_t)n * CCH * HWSZ + t;
        float s = 0.0f;
        #pragma unroll 8
        for (int c = 0; c < CCH; ++c) s += p[(size_t)c * HWSZ];
        fre[t] = s;
    }
    __syncthreads();

    if (t < HWSZ) {
        // rank in descending order, ties broken by index (stable argsort of -fre)
        const float v = fre[t];
        int rank = 0;
        for (int j = 0; j < HWSZ; ++j) {
            const float vj = fre[j];
            rank += (vj > v) || (vj == v && j < t);
        }
        sorted[rank] = t;
    }
    __syncthreads();

    if (t < 2 * KSEL) {
        const int mstart = HWSZ / 2 - KSEL / 2 - 1;   // 89
        const int pos = (t < KSEL) ? sorted[t] : sorted[mstart + (t - KSEL)];
        scene_ws[n * 2 * KSEL + t] = pos;
        const size_t base = (size_t)2 * NSAMP * KSEL * ODIM;  // after max_graph, med_graph
        out[base + (size_t)n * 2 * KSEL + t] = (float)(pos / HDIM);               // rows
        out[base + (size_t)NSAMP * 2 * KSEL + (size_t)n * 2 * KSEL + t]
            = (float)(pos % HDIM);                                               // cols
    }
}

// ---------------------------------------------------------------------------
// Kernel 2: per (sample, half): gather nodes 16x1024, GEMM vs W (1024x256)
// via V_WMMA_F32_16X16X4_F32, BN+ReLU, then L(16x16) @ y via 4 more WMMAs.
// 512 threads = 16 waves; wave w owns O-tile [16w, 16w+16).
// ---------------------------------------------------------------------------
__global__ __launch_bounds__(512) void k_graph(const float* __restrict__ fpam,
                                               const int* __restrict__ scene_ws,
                                               const float* __restrict__ w_max,
                                               const float* __restrict__ g_max,
                                               const float* __restrict__ b_max,
                                               const float* __restrict__ m_max,
                                               const float* __restrict__ v_max,
                                               const float* __restrict__ w_med,
                                               const float* __restrict__ g_med,
                                               const float* __restrict__ b_med,
                                               const float* __restrict__ m_med,
                                               const float* __restrict__ v_med,
                                               float* __restrict__ out)
{
    __shared__ float As[KSEL][260];   // node K-chunk, padded rows -> conflict-free ds_load_b64
    __shared__ float Ys[KSEL][260];   // scratch: adjacency A, later BN+ReLU output y (16 x 256)
    __shared__ float Lm[KSEL][17];    // normalized Laplacian
    __shared__ float dinv[KSEL];
    __shared__ int   ppos[KSEL];
    __shared__ int   pr[KSEL], pc[KSEL];

    const int blk  = blockIdx.x;
    const int n    = blk >> 1;
    const int half = blk & 1;
    const int t    = threadIdx.x;
    const int lane = t & 31;
    const int wave = t >> 5;
    const int lrow = lane & 15;        // M index (A) / N index (B)
    const int khi  = (lane >> 4) << 1; // 0 for lanes 0-15, 2 for lanes 16-31

    const float* W  = half ? w_med : w_max;
    const float* gP = half ? g_med : g_max;
    const float* bP = half ? b_med : b_max;
    const float* mP = half ? m_med : m_max;
    const float* vP = half ? v_med : v_max;

    // --- positions and graph Laplacian ---
    if (t < KSEL) {
        const int p = scene_ws[n * 2 * KSEL + half * KSEL + t];
        ppos[t] = p;
        pr[t] = p / HDIM;
        pc[t] = p % HDIM;
    }
    __syncthreads();
    if (t < KSEL * KSEL) {
        const int i = t >> 4, j = t & 15;
        const float dr = (float)(pr[i] - pr[j]);
        const float dc = (float)(pc[i] - pc[j]);
        Ys[i][j] = 1.0f / (1.0f + sqrtf(dr * dr + dc * dc));   // adjacency A
    }
    __syncthreads();
    if (t < KSEL) {
        float s = 0.0f;
        #pragma unroll
        for (int j = 0; j < KSEL; ++j) s += Ys[t][j];
        dinv[t] = rsqrtf(s);
    }
    __syncthreads();
    if (t < KSEL * KSEL) {
        const int i = t >> 4, j = t & 15;
        Lm[i][j] = Ys[i][j] * dinv[i] * dinv[j];
    }

    // --- per-lane BN constants (o is uniform across the 8 accum VGPRs) ---
    const int o = wave * KSEL + lrow;
    const float scale = gP[o] * rsqrtf(vP[o] + BN_EPS);
    const float shift = bP[o] - mP[o] * scale;

    // --- stage 1: y_pre = nodes(16x1024) @ W^T tile, K chunked through LDS ---
    v8f acc = {};
    const float* fbase = fpam + (size_t)n * CCH * HWSZ;
    const float* wrow0 = W + (size_t)o * CCH + khi;

    for (int kc = 0; kc < CCH; kc += 256) {
        __syncthreads();
        // cooperative gather: As[j][kk] = fpam[n, kc+kk, pos_j]
        for (int q = t; q < KSEL * 256; q += 512) {
            const int j  = q & 15;
            const int kk = q >> 4;
            As[j][kk] = fbase[(size_t)(kc + kk) * HWSZ + ppos[j]];
        }
        __syncthreads();

        const float* wrow = wrow0 + kc;
        #pragma unroll 4
        for (int kb = 0; kb < 64; ++kb) {
            const int k0 = kb * 4;
            v2f a, bb;
            a.x  = As[lrow][k0 + khi];
            a.y  = As[lrow][k0 + khi + 1];
            bb.x = wrow[k0];
            bb.y = wrow[k0 + 1];
            acc = __builtin_amdgcn_wmma_f32_16x16x4_f32(
                false, a, false, bb, (short)0, acc, false, false);
        }
    }
    __syncthreads();

    // --- BN + ReLU, stage y into LDS in row-major (j, o) for stage-2 B frags ---
    #pragma unroll
    for (int e = 0; e < 8; ++e) {
        float y = acc[e] * scale + shift;
        y = fmaxf(y, 0.0f);
        const int mrow = e + ((lane >> 4) << 3);
        Ys[mrow][o] = y;
    }
    __syncthreads();

    // --- stage 2: out = L(16x16) @ y(16x256), 4 WMMA k-steps ---
    v8f g2 = {};
    #pragma unroll
    for (int kb = 0; kb < 4; ++kb) {
        const int k0 = kb * 4 + khi;
        v2f a, bb;
        a.x  = Lm[lrow][k0];
        a.y  = Lm[lrow][k0 + 1];
        bb.x = Ys[k0][o];
        bb.y = Ys[k0 + 1][o];
        g2 = __builtin_amdgcn_wmma_f32_16x16x4_f32(
            false, a, false, bb, (short)0, g2, false, false);
    }

    // --- store: out[half][n][i][o] ---
    float* obase = out + (size_t)half * NSAMP * KSEL * ODIM
                       + (size_t)n * KSEL * ODIM + o;
    #pragma unroll
    for (int e = 0; e < 8; ++e) {
        const int i = e + ((lane >> 4) << 3);
        obase[(size_t)i * ODIM] = g2[e];
    }
}

// ---------------------------------------------------------------------------
extern "C" void kernel_launch(void* const* d_in, const int* in_sizes, int n_in,
                              void* d_out, int out_size, void* d_ws, size_t ws_size,
                              hipStream_t stream) {
    const float* fpam  = (const float*)d_in[0];
    // d_in[1] (resnet_output) is unused by the reference outputs.
    const float* w_max = (const float*)d_in[2];
    const float* g_max = (const float*)d_in[3];
    const float* b_max = (const float*)d_in[4];
    const float* m_max = (const float*)d_in[5];
    const float* v_max = (const float*)d_in[6];
    const float* w_med = (const float*)d_in[7];
    const float* g_med = (const float*)d_in[8];
    const float* b_med = (const float*)d_in[9];
    const float* m_med = (const float*)d_in[10];
    const float* v_med = (const float*)d_in[11];

    int*   scene_ws = (int*)d_ws;          // N * 32 ints
    float* out      = (float*)d_out;

    k_scene<<<NSAMP, 256, 0, stream>>>(fpam, scene_ws, out);
    k_graph<<<NSAMP * 2, 512, 0, stream>>>(fpam, scene_ws,
                                           w_max, g_max, b_max, m_max, v_max,
                                           w_med, g_med, b_med, m_med, v_med,
                                           out);
}